// GCN_81131932221570
// MI455X (gfx1250) — compile-verified
//
#include <hip/hip_runtime.h>
#include <hip/hip_bf16.h>
#include <stdint.h>

typedef __attribute__((ext_vector_type(2))) float v2f;
typedef __attribute__((ext_vector_type(8))) float v8f;

#define BATCH   16
#define WLP2    200
#define BLEN    256
#define DB      6144
#define DG      512
#define ALLD    7680      // DB + 3*DG
#define H1      1024
#define NCLS    10
#define MAXROWS (16*198)  // max sum(word_length) = 3168

__device__ __forceinline__ v8f wmma_f32(v2f a, v2f b, v8f c) {
  // D = A(16x4,f32) * B(4x16,f32) + C(16x16,f32), full f32 precision
  return __builtin_amdgcn_wmma_f32_16x16x4_f32(
      /*neg_a=*/false, a, /*neg_b=*/false, b,
      /*c_mod=*/(short)0, c, /*reuse_a=*/false, /*reuse_b=*/false);
}

// ---------------------------------------------------------------------------
// Batched WMMA GEMM: C[b] = A[b](MxK) * B[b or shared](KxN) + bias
// One wave -> 32x32 C tile (2x2 grid of 16x16 WMMA tiles): A fragments reused
// across 2 N-tiles, B fragments across 2 M-tiles -> 6 loads per 4 wmma.
// A-frag: lanes 0-15 rows m..m+15 @ K=(k,k+1); lanes 16-31 @ K=(k+2,k+3).
// B-frag: lanes 0-15 cols n..n+15 @ K=(k,k+1); lanes 16-31 @ K=(k+2,k+3).
// M-tail handled by clamping A row pointers (only masked-out D rows polluted)
// and masking C stores. N must be a multiple of 32 (true for all callers).
// ---------------------------------------------------------------------------
__global__ void wmma_gemm_f32(const float* __restrict__ A, long long sA, int lda,
                              const float* __restrict__ Bm, long long sB, int ldb,
                              const float* __restrict__ bias,
                              float* __restrict__ C, long long sC, int ldc,
                              int M, int N, int K) {
  const int b    = blockIdx.z;
  const int lane = threadIdx.x;
  const int half = lane >> 4;     // 0: K=(k,k+1)   1: K=(k+2,k+3)
  const int l    = lane & 15;
  const int n0 = (blockIdx.x * blockDim.y + threadIdx.y) * 32;
  if (n0 >= N) return;
  const int m0 = blockIdx.y * 32;

  const float* Ab = A  + (long long)b * sA;
  const float* Bb = Bm + (long long)b * sB;

  int r0 = m0 + l;       if (r0 > M - 1) r0 = M - 1;   // clamp: stores masked below
  int r1 = m0 + 16 + l;  if (r1 > M - 1) r1 = M - 1;
  const float* a0p = Ab + (long long)r0 * lda + 2 * half;
  const float* a1p = Ab + (long long)r1 * lda + 2 * half;
  const float* bp0 = Bb + (long long)(2 * half) * ldb + n0 + l;
  const float* bp1 = bp0 + 16;

  v8f acc00 = {}, acc01 = {}, acc10 = {}, acc11 = {};
  #pragma unroll 2
  for (int k = 0; k < K; k += 4) {
    v2f a0 = *(const v2f*)(a0p + k);          // global_load_b64
    v2f a1 = *(const v2f*)(a1p + k);
    v2f bf0, bf1;
    bf0.x = bp0[(long long)k * ldb];  bf0.y = bp0[(long long)(k + 1) * ldb];
    bf1.x = bp1[(long long)k * ldb];  bf1.y = bp1[(long long)(k + 1) * ldb];
    acc00 = wmma_f32(a0, bf0, acc00);
    acc01 = wmma_f32(a0, bf1, acc01);
    acc10 = wmma_f32(a1, bf0, acc10);
    acc11 = wmma_f32(a1, bf1, acc11);
  }

  const float bv0 = bias ? bias[n0 + l]      : 0.0f;
  const float bv1 = bias ? bias[n0 + 16 + l] : 0.0f;
  float* Cb = C + (long long)b * sC;
  #pragma unroll
  for (int r = 0; r < 8; ++r) {
    const int row0 = m0 + r + half * 8;        // D layout: VGPR r = M row r / 8+r
    const int row1 = row0 + 16;
    if (row0 < M) {
      Cb[(long long)row0 * ldc + n0 + l]      = acc00[r] + bv0;
      Cb[(long long)row0 * ldc + n0 + 16 + l] = acc01[r] + bv1;
    }
    if (row1 < M) {
      Cb[(long long)row1 * ldc + n0 + l]      = acc10[r] + bv0;
      Cb[(long long)row1 * ldc + n0 + 16 + l] = acc11[r] + bv1;
    }
  }
}

// ---------------------------------------------------------------------------
// Ragged row -> (batch,pos) base offset via prefix sum of word_length.
// Rows past the total clamp to base 0 (valid memory; stores are masked).
// ---------------------------------------------------------------------------
__device__ __forceinline__ long long map_row(const int* __restrict__ wl,
                                             int row, int* totalOut) {
  int total = 0, bsel = 0, pos = 0, found = 0;
  #pragma unroll
  for (int b = 0; b < BATCH; ++b) {
    int w = wl[b];
    if (!found && row < total + w) { found = 1; bsel = b; pos = row - total + 1; }
    total += w;
  }
  *totalOut = total;
  return found ? ((long long)bsel * WLP2 + pos) : 0LL;
}

// ---------------------------------------------------------------------------
// FC1: h[row] = tanh(clean_reps[row] @ W_fc1 + b_fc1), 32x32 tile per wave.
// The concat is handled as 4 branch-free segment loops (all 4-aligned).
// ---------------------------------------------------------------------------
#define FC1_SEG(P0, P1, LEN)                                              \
  for (int j = 0; j < (LEN); j += 4) {                                    \
    v2f a0 = *(const v2f*)((P0) + j);                                     \
    v2f a1 = *(const v2f*)((P1) + j);                                     \
    v2f bf0, bf1;                                                         \
    bf0.x = bp0[(long long)j * H1]; bf0.y = bp0[(long long)(j + 1) * H1]; \
    bf1.x = bp1[(long long)j * H1]; bf1.y = bp1[(long long)(j + 1) * H1]; \
    acc00 = wmma_f32(a0, bf0, acc00);                                     \
    acc01 = wmma_f32(a0, bf1, acc01);                                     \
    acc10 = wmma_f32(a1, bf0, acc10);                                     \
    acc11 = wmma_f32(a1, bf1, acc11);                                     \
  }                                                                       \
  bp0 += (long long)(LEN) * H1; bp1 += (long long)(LEN) * H1;

__global__ void fc1_tanh_kernel(const float* __restrict__ emb, const float* __restrict__ xb,
                                const float* __restrict__ o1,  const float* __restrict__ o2,
                                const int* __restrict__ wl,
                                const float* __restrict__ W,   const float* __restrict__ bias,
                                float* __restrict__ h) {
  const int lane = threadIdx.x;
  const int half = lane >> 4;
  const int l    = lane & 15;
  const int n0 = (blockIdx.x * blockDim.y + threadIdx.y) * 32;
  const int m0 = blockIdx.y * 32;
  const int koff = 2 * half;

  int total;
  const long long rb0 = map_row(wl, m0 + l,      &total);
  const long long rb1 = map_row(wl, m0 + 16 + l, &total);

  const float* bp0 = W + (long long)koff * H1 + n0 + l;
  const float* bp1 = bp0 + 16;

  v8f acc00 = {}, acc01 = {}, acc10 = {}, acc11 = {};

  { const float* p0 = emb + rb0 * DB + koff;
    const float* p1 = emb + rb1 * DB + koff;
    FC1_SEG(p0, p1, DB) }
  { const float* p0 = xb + rb0 * DG + koff;
    const float* p1 = xb + rb1 * DG + koff;
    FC1_SEG(p0, p1, DG) }
  { const float* p0 = o1 + rb0 * DG + koff;
    const float* p1 = o1 + rb1 * DG + koff;
    FC1_SEG(p0, p1, DG) }
  { const float* p0 = o2 + rb0 * DG + koff;
    const float* p1 = o2 + rb1 * DG + koff;
    FC1_SEG(p0, p1, DG) }

  const float bv0 = bias[n0 + l];
  const float bv1 = bias[n0 + 16 + l];
  #pragma unroll
  for (int r = 0; r < 8; ++r) {
    const int row0 = m0 + r + half * 8;
    const int row1 = row0 + 16;
    if (row0 < total) {
      h[(long long)row0 * H1 + n0 + l]      = tanhf(acc00[r] + bv0);
      h[(long long)row0 * H1 + n0 + 16 + l] = tanhf(acc01[r] + bv1);
    }
    if (row1 < total) {
      h[(long long)row1 * H1 + n0 + l]      = tanhf(acc10[r] + bv0);
      h[(long long)row1 * H1 + n0 + 16 + l] = tanhf(acc11[r] + bv1);
    }
  }
}

// ---------------------------------------------------------------------------
// FC2 + argmax: one wave per packed row. logits -> out[0 .. total*10),
// preds (as float) -> out[total*10 .. total*11).
// ---------------------------------------------------------------------------
__global__ void fc2_argmax_kernel(const float* __restrict__ h, const int* __restrict__ wl,
                                  const float* __restrict__ W2, const float* __restrict__ b2,
                                  float* __restrict__ out) {
  const int r = blockIdx.x;
  const int lane = threadIdx.x;
  int total = 0;
  #pragma unroll
  for (int b = 0; b < BATCH; ++b) total += wl[b];
  if (r >= total) return;

  const float* hr = h + (long long)r * H1;
  float best = -3.402823e38f; int bestc = 0;
  for (int c = 0; c < NCLS; ++c) {
    float s = 0.0f;
    for (int k = lane; k < H1; k += 32) s += hr[k] * W2[k * NCLS + c];
    #pragma unroll
    for (int off = 16; off > 0; off >>= 1) s += __shfl_xor(s, off, 32);
    s += b2[c];
    if (lane == 0) {
      out[(long long)r * NCLS + c] = s;
      if (s > best) { best = s; bestc = c; }  // strict > keeps first max (jnp.argmax)
    }
  }
  if (lane == 0) out[(long long)total * NCLS + r] = (float)bestc;
}

// ---------------------------------------------------------------------------
extern "C" void kernel_launch(void* const* d_in, const int* in_sizes, int n_in,
                              void* d_out, int out_size, void* d_ws, size_t ws_size,
                              hipStream_t stream) {
  const float* bert_x    = (const float*)d_in[0];
  const float* transform = (const float*)d_in[1];
  const float* adj       = (const float*)d_in[2];
  const int*   wl        = (const int*)  d_in[3];
  const float* W_prj = (const float*)d_in[4];
  const float* b_prj = (const float*)d_in[5];
  const float* W_gc1 = (const float*)d_in[6];
  const float* b_gc1 = (const float*)d_in[7];
  const float* W_gc2 = (const float*)d_in[8];
  const float* b_gc2 = (const float*)d_in[9];
  const float* W_fc1 = (const float*)d_in[10];
  const float* b_fc1 = (const float*)d_in[11];
  const float* W_fc2 = (const float*)d_in[12];
  const float* b_fc2 = (const float*)d_in[13];
  float* out = (float*)d_out;

  // workspace layout (floats)
  float* emb = (float*)d_ws;                          // [16,200,6144]
  float* xb  = emb + (size_t)BATCH * WLP2 * DB;       // [16,200,512]
  float* t1  = xb  + (size_t)BATCH * WLP2 * DG;       // temp x@Wgc  [16,200,512]
  float* o1  = t1  + (size_t)BATCH * WLP2 * DG;       // [16,200,512]
  float* o2  = o1  + (size_t)BATCH * WLP2 * DG;       // [16,200,512]
  float* h   = o2  + (size_t)BATCH * WLP2 * DG;       // [3168,1024]

  const dim3 blk(32, 4, 1);
  const int mT = (WLP2 + 31) / 32;  // 7 row-blocks of 32

  // 1) embeddings = transform @ bert_x                 (200x256)*(256x6144)
  { dim3 g(DB / 128, mT, BATCH);
    hipLaunchKernelGGL(wmma_gemm_f32, g, blk, 0, stream,
      transform, (long long)WLP2*BLEN, BLEN, bert_x, (long long)BLEN*DB, DB,
      (const float*)nullptr, emb, (long long)WLP2*DB, DB, WLP2, DB, BLEN); }

  // 2) x = embeddings @ W_prj + b_prj                  (200x6144)*(6144x512)
  { dim3 g(DG / 128, mT, BATCH);
    hipLaunchKernelGGL(wmma_gemm_f32, g, blk, 0, stream,
      emb, (long long)WLP2*DB, DB, W_prj, 0LL, DG,
      b_prj, xb, (long long)WLP2*DG, DG, WLP2, DG, DB); }

  // 3) t1 = x @ W_gc1
  { dim3 g(DG / 128, mT, BATCH);
    hipLaunchKernelGGL(wmma_gemm_f32, g, blk, 0, stream,
      xb, (long long)WLP2*DG, DG, W_gc1, 0LL, DG,
      (const float*)nullptr, t1, (long long)WLP2*DG, DG, WLP2, DG, DG); }

  // 4) out1 = adj @ t1 + b_gc1                         (200x200)*(200x512)
  { dim3 g(DG / 128, mT, BATCH);
    hipLaunchKernelGGL(wmma_gemm_f32, g, blk, 0, stream,
      adj, (long long)WLP2*WLP2, WLP2, t1, (long long)WLP2*DG, DG,
      b_gc1, o1, (long long)WLP2*DG, DG, WLP2, DG, WLP2); }

  // 5) t1 = out1 @ W_gc2
  { dim3 g(DG / 128, mT, BATCH);
    hipLaunchKernelGGL(wmma_gemm_f32, g, blk, 0, stream,
      o1, (long long)WLP2*DG, DG, W_gc2, 0LL, DG,
      (const float*)nullptr, t1, (long long)WLP2*DG, DG, WLP2, DG, DG); }

  // 6) out2 = adj @ t1 + b_gc2
  { dim3 g(DG / 128, mT, BATCH);
    hipLaunchKernelGGL(wmma_gemm_f32, g, blk, 0, stream,
      adj, (long long)WLP2*WLP2, WLP2, t1, (long long)WLP2*DG, DG,
      b_gc2, o2, (long long)WLP2*DG, DG, WLP2, DG, WLP2); }

  // 7) h = tanh(gathered_concat @ W_fc1 + b_fc1)       (<=3168 x 7680)*(7680x1024)
  { dim3 g(H1 / 128, MAXROWS / 32, 1);
    hipLaunchKernelGGL(fc1_tanh_kernel, g, blk, 0, stream,
      emb, xb, o1, o2, wl, W_fc1, b_fc1, h); }

  // 8) logits + argmax
  hipLaunchKernelGGL(fc2_argmax_kernel, dim3(MAXROWS), dim3(32), 0, stream,
    h, wl, W_fc2, b_fc2, out);
}